// VitsEncoder_11304353923185
// MI455X (gfx1250) — compile-verified
//
#include <hip/hip_runtime.h>
#include <math.h>

// Problem constants (from reference): B=8, C=192, TX=512, T_MAX=4096
#define BATCH   8
#define CHANS   192
#define TXLEN   512
#define TYLEN   4096
#define FTILE   1024  // frames per expand block (256 threads x 4 frames)
#define CGRP    12    // channels per expand block (192 / 12 = 16 groups)

typedef int v4i __attribute__((ext_vector_type(4)));

// ---------------------------------------------------------------------------
// Kernel 1: durations -> cumulative path -> y_mask + frame->phoneme index map
// One block per batch, 512 threads (one per phoneme). Tiny kernel.
// ---------------------------------------------------------------------------
__global__ __launch_bounds__(TXLEN) void vits_prep_kernel(
    const float* __restrict__ logw,      // [B,1,TX]
    const float* __restrict__ scales,    // [3]  (noise_scale, length_scale, _)
    const int*   __restrict__ x_lengths, // [B]
    float*       __restrict__ y_mask,    // [B,1,TY] (output tail)
    int*         __restrict__ idx)       // [B,TY] workspace: phoneme id or -1
{
    const int b = blockIdx.x;
    const int t = threadIdx.x;           // phoneme index tx

    __shared__ float s[TXLEN];

    const float length_scale = scales[1];
    int xl = x_lengths[b];
    xl = xl < 1 ? 1 : (xl > TXLEN ? TXLEN : xl);

    const float msk = (t < xl) ? 1.0f : 0.0f;
    const float w   = expf(logw[b * TXLEN + t]) * msk * length_scale;
    const float wc  = ceilf(w);          // integer-valued float >= 0

    // Inclusive Hillis-Steele scan over 512 phonemes in LDS
    s[t] = wc;
    __syncthreads();
#pragma unroll
    for (int off = 1; off < TXLEN; off <<= 1) {
        const float add = (t >= off) ? s[t - off] : 0.0f;
        __syncthreads();
        s[t] += add;
        __syncthreads();
    }

    const float total = s[TXLEN - 1];
    const float ylen  = fminf(fmaxf(total, 1.0f), (float)TYLEN);

    // y_mask[b, ty] = (ty < y_length)
    for (int ty = t; ty < TYLEN; ty += TXLEN)
        y_mask[b * TYLEN + ty] = ((float)ty < ylen) ? 1.0f : 0.0f;

    // init index map to -1 (frames beyond duration keep z = noise*noise_scale)
    for (int ty = t; ty < TYLEN; ty += TXLEN)
        idx[b * TYLEN + ty] = -1;
    __syncthreads();  // init visible to all scatter writers in this block

    // scatter: frames [cum-wc, min(cum, ylen)) belong to phoneme t
    const float cumv   = s[t];
    const float startf = cumv - wc;
    const int si = (int)fminf(startf, ylen);
    const int ei = (int)fminf(cumv,  ylen);
    for (int ty = si; ty < ei; ++ty)
        idx[b * TYLEN + ty] = t;
}

// ---------------------------------------------------------------------------
// Kernel 2: z_p[b,c,ty] = m_p[b,c,tx] + noise[b,c,ty]*exp(logs_p[b,c,tx])*ns
// (tx = idx[b][ty]; tx < 0 -> gathered stats are 0 -> z = noise*ns)
// Each thread owns 4 consecutive frames: noise/z move as B128 vectors.
// Index tile staged into LDS via CDNA5 async-to-LDS (ASYNCcnt-tracked).
// grid = (B*TY/FTILE, C/CGRP) = (32, 16); 256 threads/block.
// ---------------------------------------------------------------------------
__global__ __launch_bounds__(256) void vits_expand_kernel(
    const float* __restrict__ m_p,     // [B,C,TX]
    const float* __restrict__ logs_p,  // [B,C,TX]
    const float* __restrict__ noise,   // [B,C,TY]
    const float* __restrict__ scales,  // [3]
    const int*   __restrict__ idx,     // [B,TY]
    float*       __restrict__ z)       // [B,C,TY]
{
    const int tileId = blockIdx.x;             // 0..31
    const int b      = tileId >> 2;            // 4 tiles of 1024 frames
    const int ty0    = (tileId & 3) * FTILE;
    const int cbase  = blockIdx.y * CGRP;
    const int t      = threadIdx.x;            // owns frames ty0+4t..ty0+4t+3

    __shared__ int sidx[FTILE];

    const int* gsrc = idx + b * TYLEN + ty0 + 4 * t;   // 16B-aligned

#if __has_builtin(__builtin_amdgcn_global_load_async_to_lds_b128)
    __builtin_amdgcn_global_load_async_to_lds_b128(
        (__attribute__((address_space(1))) v4i*)(v4i*)gsrc,
        (__attribute__((address_space(3))) v4i*)(v4i*)&sidx[4 * t],
        0, 0);
  #if __has_builtin(__builtin_amdgcn_s_wait_asynccnt)
    __builtin_amdgcn_s_wait_asynccnt(0);
  #else
    asm volatile("s_wait_asynccnt 0" ::: "memory");
  #endif
#elif __has_builtin(__builtin_amdgcn_global_load_async_to_lds_b32)
    #pragma unroll
    for (int k = 0; k < 4; ++k)
        __builtin_amdgcn_global_load_async_to_lds_b32(
            (__attribute__((address_space(1))) int*)(int*)(gsrc + k),
            (__attribute__((address_space(3))) int*)&sidx[4 * t + k],
            0, 0);
  #if __has_builtin(__builtin_amdgcn_s_wait_asynccnt)
    __builtin_amdgcn_s_wait_asynccnt(0);
  #else
    asm volatile("s_wait_asynccnt 0" ::: "memory");
  #endif
#else
    *(int4*)&sidx[4 * t] = *(const int4*)gsrc;
#endif
    __syncthreads();

    const float ns  = scales[0];
    const int   ty  = ty0 + 4 * t;
    const int4  txv = *(const int4*)&sidx[4 * t];      // ds_load_b128

#pragma unroll 4
    for (int ci = 0; ci < CGRP; ++ci) {
        const int    c  = cbase + ci;
        const size_t pb = ((size_t)(b * CHANS + c)) * TXLEN;
        const size_t fb = ((size_t)(b * CHANS + c)) * TYLEN + (size_t)ty;

        if (ci + 1 < CGRP)  // hint next channel's frame line toward L2
            __builtin_prefetch(noise + fb + TYLEN, 0, 0);

        const float4 nv = *(const float4*)&noise[fb];  // global_load_b128

        // branchless per-frame gather: clamp index, load, select identity
        auto comp = [&](int tx, float n) -> float {
            const int   txs = tx < 0 ? 0 : tx;
            float m = m_p[pb + (size_t)txs];
            float e = expf(logs_p[pb + (size_t)txs]);
            if (tx < 0) { m = 0.0f; e = 1.0f; }        // v_cndmask
            return m + n * e * ns;
        };

        float4 out;
        out.x = comp(txv.x, nv.x);
        out.y = comp(txv.y, nv.y);
        out.z = comp(txv.z, nv.z);
        out.w = comp(txv.w, nv.w);

        *(float4*)&z[fb] = out;                        // global_store_b128
    }
}

// ---------------------------------------------------------------------------
// Launch: inputs in setup_inputs() order:
//  0 logw [8,1,512] f32 | 1 m_p [8,192,512] f32 | 2 logs_p [8,192,512] f32
//  3 noise [8,192,4096] f32 | 4 scales [3] f32 | 5 x_lengths [8] int
// Output: z_p [8,192,4096] f32 ++ y_mask [8,1,4096] f32
// Workspace: idx map, B*TY*4 = 128 KB
// ---------------------------------------------------------------------------
extern "C" void kernel_launch(void* const* d_in, const int* in_sizes, int n_in,
                              void* d_out, int out_size, void* d_ws, size_t ws_size,
                              hipStream_t stream) {
    const float* logw      = (const float*)d_in[0];
    const float* m_p       = (const float*)d_in[1];
    const float* logs_p    = (const float*)d_in[2];
    const float* noise     = (const float*)d_in[3];
    const float* scales    = (const float*)d_in[4];
    const int*   x_lengths = (const int*)d_in[5];

    float* z      = (float*)d_out;                                  // [B,C,TY]
    float* y_mask = z + (size_t)BATCH * CHANS * TYLEN;              // [B,1,TY]
    int*   idx    = (int*)d_ws;                                     // [B,TY]

    vits_prep_kernel<<<BATCH, TXLEN, 0, stream>>>(logw, scales, x_lengths,
                                                  y_mask, idx);

    dim3 grid(BATCH * (TYLEN / FTILE), CHANS / CGRP);               // 32 x 16
    vits_expand_kernel<<<grid, 256, 0, stream>>>(m_p, logs_p, noise,
                                                 scales, idx, z);
}